// cKrausOperator_88527865905689
// MI455X (gfx1250) — compile-verified
//
#include <hip/hip_runtime.h>

// ---------------------------------------------------------------------------
// Kraus-sequence log-likelihood on MI455X (gfx1250), fp32 WMMA 16x16x4.
//
// Math: logp = -log2 tr(P rho0 P^H rhol) + L*log2(lam),  P = M_L ... M_1
// (telescoped renormalized scan). P is built as a parallel product tree with
// per-level 2^e rescaling (log2 scales accumulate additively).
// Complex matrices stored as [re plane | im plane], row-major, D=128.
//
// GEMM core: one wave per 16x16 tile, 4 real WMMA accumulators per complex
// tile. B panel staged to LDS with double-buffered async global->LDS copies
// (ASYNCcnt) when the toolchain exposes the gfx1250 builtins.
// ---------------------------------------------------------------------------

typedef __attribute__((ext_vector_type(2))) float v2f;
typedef __attribute__((ext_vector_type(8))) float v8f;
typedef int vec4i __attribute__((vector_size(16)));   // matches builtin proto

#define D      128
#define PL     (D * D)         // 16384 floats per plane
#define MATF   (2 * PL)        // floats per complex matrix
#define NKR    8
#define SEQL   8192
#define CHUNK  256
#define NCHUNK (SEQL / CHUNK)  // 32
#define PAIRS_PER_CHUNK (CHUNK / 2)  // 128
#define NPAIRS (SEQL / 2)      // 4096
#define PITERS 200

// K-chunking of the GEMM inner dimension for LDS staging
#define KT     16              // K values per staged chunk
#define KCHN   (D / KT)        // 8 chunks per matrix
#define BPAD   136             // padded LDS row (floats): disjoint bank ranges
                               // for the two 16-lane halves (K vs K+2 rows)

#if __has_builtin(__builtin_amdgcn_global_load_async_to_lds_b128) && \
    __has_builtin(__builtin_amdgcn_s_wait_asynccnt)
#define USE_ASYNC_LDS 1
#else
#define USE_ASYNC_LDS 0
#endif

// ---------------------------------------------------------------------------
// Complex GEMM: C_batch = sum_{i<n_inner} A_i * B_i   (optionally += old C)
// grid = (8, nbatch), block = 256 (8 waves); wave w computes tile
// (tm = blockIdx.x, tn = w) of the 8x8 tile grid of the 128x128 output.
// Index of A for (batch,i): (a_sel ? a_sel[batch] : batch*a_mul + a_off) + i.
// ---------------------------------------------------------------------------
__global__ __launch_bounds__(256) void k_cgemm(
    const float* __restrict__ Ab, const float* __restrict__ Bb,
    float* __restrict__ Cb,
    const int* __restrict__ a_sel, int a_mul, int a_off,
    const int* __restrict__ b_sel, int b_mul, int b_off,
    int c_mul, int c_off, int n_inner, int acc)
{
  // two async double buffers: [buf][plane*KT + row][BPAD]
  __shared__ float ldsB[2][2 * KT * BPAD];

  const int wave = threadIdx.x >> 5;
  const int lane = threadIdx.x & 31;
  const int tm   = blockIdx.x * 16;             // output row base (same all waves)
  const int tn   = wave * 16;                   // output col base (per wave)
  const int batch = blockIdx.y;

  const int ai0 = a_sel ? a_sel[batch] : batch * a_mul + a_off;
  const int bi0 = b_sel ? b_sel[batch] : batch * b_mul + b_off;
  const int half = lane >> 4;                   // 0: K pair 0, 1: K pair 1
  const int mr   = lane & 15;

  const int NC = n_inner * KCHN;                // total K chunks to stream

  // Stage B panel chunk c (16 K-rows x 128 cols x {re,im}) into buffer `buf`.
  // Per-thread: 4 x 16-byte segments; the panel is plane-contiguous in global.
  auto issueB = [&](int c, int buf) {
    const float* __restrict__ B = Bb + (long)(bi0 + (c >> 3)) * MATF;
    const int k0 = (c & (KCHN - 1)) * KT;
    for (int j = 0; j < 4; ++j) {
      const int q     = threadIdx.x + 256 * j;  // 0..1023 float4 segments
      const int plane = q >> 9;                 // 0: re, 1: im
      const int rem   = q & 511;
      const int row   = rem >> 5;               // 0..15
      const int c4    = (rem & 31) * 4;         // col (floats)
      const float* g  = B + plane * PL + (k0 + row) * D + c4;
      float* l        = &ldsB[buf][(plane * KT + row) * BPAD + c4];
#if USE_ASYNC_LDS
      __builtin_amdgcn_global_load_async_to_lds_b128(
          (__attribute__((address_space(1))) vec4i*)g,
          (__attribute__((address_space(3))) vec4i*)l, 0, 0);
#else
      *(float4*)l = *(const float4*)g;
#endif
    }
  };

  v8f c_rr = {}; v8f c_ii = {}; v8f c_ri = {}; v8f c_ir = {};

  issueB(0, 0);
  for (int c = 0; c < NC; ++c) {
#if USE_ASYNC_LDS
    __builtin_amdgcn_s_wait_asynccnt(0);
#endif
    __syncthreads();                            // buffer c&1 visible to all
    if (c + 1 < NC) issueB(c + 1, (c + 1) & 1); // overlap copy with compute

    const float* __restrict__ A = Ab + (long)(ai0 + (c >> 3)) * MATF;
    const int k0 = (c & (KCHN - 1)) * KT;
    const float* __restrict__ lb = &ldsB[c & 1][0];

    for (int kl = 0; kl < KT; kl += 4) {
      const int ka  = k0 + kl + 2 * half;       // global K for A fragment
      const int kll = kl + 2 * half;            // chunk-local K for B fragment
      // A fragment (16x4): row tm+mr, K = ka, ka+1 (contiguous b64 pairs)
      const float* ap = A + (tm + mr) * D + ka;
      v2f ar = { ap[0],  ap[1]      };
      v2f ai = { ap[PL], ap[PL + 1] };
      // B fragment (4x16) from LDS: col tn+mr, rows kll, kll+1
      const int bo = kll * BPAD + tn + mr;
      v2f br = { lb[bo],             lb[bo + BPAD] };
      v2f bi = { lb[bo + KT * BPAD], lb[bo + (KT + 1) * BPAD] };
      c_rr = __builtin_amdgcn_wmma_f32_16x16x4_f32(false, ar, false, br,
                                                   (short)0, c_rr, false, false);
      c_ii = __builtin_amdgcn_wmma_f32_16x16x4_f32(false, ai, false, bi,
                                                   (short)0, c_ii, false, false);
      c_ri = __builtin_amdgcn_wmma_f32_16x16x4_f32(false, ar, false, bi,
                                                   (short)0, c_ri, false, false);
      c_ir = __builtin_amdgcn_wmma_f32_16x16x4_f32(false, ai, false, br,
                                                   (short)0, c_ir, false, false);
    }
    __syncthreads();                            // all reads done before refill
  }

  float* __restrict__ C = Cb + (long)(batch * c_mul + c_off) * MATF;
  // C/D layout: VGPR r, lanes 0-15 -> M = r, lanes 16-31 -> M = r + 8
  for (int r = 0; r < 8; ++r) {
    const int row = tm + r + 8 * half;
    const int off = row * D + tn + mr;
    float cre = c_rr[r] - c_ii[r];
    float cim = c_ri[r] + c_ir[r];
    if (acc) { cre += C[off]; cim += C[off + PL]; }
    C[off]      = cre;
    C[off + PL] = cim;
  }
}

// Pack [8,128,128] re + [8,128,128] im into per-matrix [re|im] planes.
__global__ __launch_bounds__(256) void k_pack(
    const float* __restrict__ re, const float* __restrict__ im,
    float* __restrict__ dst)
{
  int idx = blockIdx.x * 256 + threadIdx.x;   // < 8*PL
  int a = idx >> 14;
  int r = idx & (PL - 1);
  dst[(long)a * MATF + r]      = re[idx];
  dst[(long)a * MATF + PL + r] = im[idx];
}

// Conjugate transpose: dst = src^H ; grid = (PL/256, nbatch)
__global__ __launch_bounds__(256) void k_conjT(
    const float* __restrict__ src, float* __restrict__ dst)
{
  int b = blockIdx.y;
  const float* S = src + (long)b * MATF;
  float*       T = dst + (long)b * MATF;
  int idx = blockIdx.x * 256 + threadIdx.x;   // < PL
  int i = idx >> 7, j = idx & (D - 1);
  T[j * D + i]      =  S[idx];
  T[j * D + i + PL] = -S[idx + PL];
}

__global__ __launch_bounds__(256) void k_init_eye(float* __restrict__ m, float scale)
{
  int idx = blockIdx.x * 256 + threadIdx.x;   // < PL
  int i = idx >> 7, j = idx & (D - 1);
  m[idx]      = (i == j) ? scale : 0.0f;
  m[idx + PL] = 0.0f;
}

// dst = src / tr(src) ; optionally store tr (the power-iter eigenvalue mu).
__global__ __launch_bounds__(256) void k_trace_scale(
    const float* __restrict__ src, float* __restrict__ dst, float* mu_out)
{
  __shared__ float red[256];
  float s = (threadIdx.x < D) ? src[threadIdx.x * D + threadIdx.x] : 0.0f;
  red[threadIdx.x] = s;
  __syncthreads();
  for (int t = 128; t > 0; t >>= 1) {
    if (threadIdx.x < t) red[threadIdx.x] += red[threadIdx.x + t];
    __syncthreads();
  }
  const float inv = 1.0f / red[0];
  for (int idx = threadIdx.x; idx < MATF; idx += 256) dst[idx] = src[idx] * inv;
  if (threadIdx.x == 0 && mu_out) mu_out[0] = red[0];
}

// out = Re tr(A*B) = sum_ij Ar[i,j]Br[j,i] - Ai[i,j]Bi[j,i]
__global__ __launch_bounds__(256) void k_dot_trace(
    const float* __restrict__ A, const float* __restrict__ B, float* out)
{
  __shared__ float red[256];
  float acc = 0.0f;
  for (int idx = threadIdx.x; idx < PL; idx += 256) {
    int i = idx >> 7, j = idx & (D - 1);
    int tj = j * D + i;
    acc += A[idx] * B[tj] - A[idx + PL] * B[tj + PL];
  }
  red[threadIdx.x] = acc;
  __syncthreads();
  for (int t = 128; t > 0; t >>= 1) {
    if (threadIdx.x < t) red[threadIdx.x] += red[threadIdx.x + t];
    __syncthreads();
  }
  if (threadIdx.x == 0) out[0] = red[0];
}

__global__ __launch_bounds__(256) void k_scale_inv(
    float* __restrict__ m, const float* __restrict__ s)
{
  int idx = blockIdx.x * 256 + threadIdx.x;   // < MATF
  m[idx] = m[idx] / s[0];
}

// Per-matrix power-of-two rescale (keeps products in fp32 range) with
// additive log2-scale bookkeeping: lo[b] = e + la[b*la_mul+la_off] + lb[...].
// One block per matrix.
__global__ __launch_bounds__(256) void k_rescale(
    float* __restrict__ base,
    const float* __restrict__ la, int la_mul, int la_off,
    const float* __restrict__ lb, int lb_mul, int lb_off,
    float* __restrict__ lo)
{
  int b = blockIdx.x;
  float* M = base + (long)b * MATF;
  __shared__ float red[256];
  float mx = 0.0f;
  for (int idx = threadIdx.x; idx < MATF; idx += 256) mx = fmaxf(mx, fabsf(M[idx]));
  red[threadIdx.x] = mx;
  __syncthreads();
  for (int t = 128; t > 0; t >>= 1) {
    if (threadIdx.x < t) red[threadIdx.x] = fmaxf(red[threadIdx.x], red[threadIdx.x + t]);
    __syncthreads();
  }
  const float m = red[0];
  const float e = (m > 0.0f) ? floorf(log2f(m)) : 0.0f;
  const float s = exp2f(-e);
  for (int idx = threadIdx.x; idx < MATF; idx += 256) M[idx] *= s;
  if (threadIdx.x == 0) {
    float add = e;
    if (la) add += la[b * la_mul + la_off];
    if (lb) add += lb[b * lb_mul + lb_off];
    lo[b] = add;
  }
}

__global__ void k_set_scalar(float* p, float v) { p[0] = v; }

// Pair-gather of the symbol sequence: selA[p] = seq[2p+1], selB[p] = seq[2p].
__global__ __launch_bounds__(256) void k_make_sel(
    const int* __restrict__ seq, int* __restrict__ selA, int* __restrict__ selB)
{
  int p = blockIdx.x * 256 + threadIdx.x;   // < NPAIRS
  selA[p] = seq[2 * p + 1];
  selB[p] = seq[2 * p];
}

// logp = -(log2(v) + Slog) + L * log2(lam)
__global__ void k_finalize(const float* v, const float* Slog,
                           const float* lam, float* out, float Lf)
{
  out[0] = -(log2f(v[0]) + Slog[0]) + Lf * log2f(lam[0]);
}

// ---------------------------------------------------------------------------
extern "C" void kernel_launch(void* const* d_in, const int* in_sizes, int n_in,
                              void* d_out, int out_size, void* d_ws, size_t ws_size,
                              hipStream_t stream)
{
  (void)in_sizes; (void)n_in; (void)out_size; (void)ws_size;
  const float* Kre = (const float*)d_in[0];
  const float* Kim = (const float*)d_in[1];
  const int*   seq = (const int*)d_in[2];

  // -------- workspace carve (all L2-resident: ~30 MB) --------
  float* w    = (float*)d_ws;
  float* KC   = w; w += (long)NKR * MATF;     // packed K
  float* KH   = w; w += (long)NKR * MATF;     // K^H
  float* T    = w; w += (long)NKR * MATF;     // per-op temporaries
  float* RHOV = w; w += MATF;
  float* RHOR = w; w += MATF;                 // becomes rho0
  float* RHOL = w; w += MATF;
  float* Pm[2]; Pm[0] = w; w += MATF; Pm[1] = w; w += MATF;
  float* PH   = w; w += MATF;
  float* W1   = w; w += MATF;
  float* W2   = w; w += MATF;
  float* LVA  = w; w += (long)PAIRS_PER_CHUNK * MATF;       // 128 matrices
  float* LVB  = w; w += (long)(PAIRS_PER_CHUNK / 2) * MATF; // 64 matrices
  float* logsA = w; w += PAIRS_PER_CHUNK;
  float* logsB = w; w += PAIRS_PER_CHUNK / 2;
  float* Plog[2]; Plog[0] = w; w += 1; Plog[1] = w; w += 1;
  float* lam  = w; w += 1;
  float* tval = w; w += 1;
  float* vval = w; w += 1;
  int* selA = (int*)w;
  int* selB = selA + NPAIRS;

  // -------- setup --------
  k_pack<<<(NKR * PL) / 256, 256, 0, stream>>>(Kre, Kim, KC);
  k_conjT<<<dim3(PL / 256, NKR), 256, 0, stream>>>(KC, KH);
  k_init_eye<<<PL / 256, 256, 0, stream>>>(RHOR, 1.0f / (float)D);
  k_init_eye<<<PL / 256, 256, 0, stream>>>(RHOL, 1.0f);
  k_make_sel<<<NPAIRS / 256, 256, 0, stream>>>(seq, selA, selB);

  // -------- power iteration: right eigenmatrix --------
  for (int it = 0; it < PITERS; ++it) {
    // T_a = K_a * rho_r   (8 parallel complex GEMMs)
    k_cgemm<<<dim3(8, NKR), 256, 0, stream>>>(KC, RHOR, T,
        nullptr, 1, 0, nullptr, 0, 0, 1, 0, 1, 0);
    // rho_v = sum_a T_a * KH_a   (8-way accumulating GEMM, one batch)
    k_cgemm<<<dim3(8, 1), 256, 0, stream>>>(T, KH, RHOV,
        nullptr, 0, 0, nullptr, 0, 0, 0, 0, NKR, 0);
    k_trace_scale<<<1, 256, 0, stream>>>(RHOV, RHOR, nullptr);
  }
  // -------- power iteration: left eigenmatrix (+ eigenvalue lam) --------
  for (int it = 0; it < PITERS; ++it) {
    k_cgemm<<<dim3(8, NKR), 256, 0, stream>>>(KH, RHOL, T,
        nullptr, 1, 0, nullptr, 0, 0, 1, 0, 1, 0);
    k_cgemm<<<dim3(8, 1), 256, 0, stream>>>(T, KC, RHOV,
        nullptr, 0, 0, nullptr, 0, 0, 0, 0, NKR, 0);
    k_trace_scale<<<1, 256, 0, stream>>>(RHOV, RHOL, lam);  // last mu -> lam
  }

  // rho0 = rho_r / Re tr(rho_r * rho_l)
  k_dot_trace<<<1, 256, 0, stream>>>(RHOR, RHOL, tval);
  k_scale_inv<<<MATF / 256, 256, 0, stream>>>(RHOR, tval);

  // -------- running product P over 32 chunks, each a 256-step tree --------
  int pc = 0;
  k_init_eye<<<PL / 256, 256, 0, stream>>>(Pm[pc], 1.0f);
  k_set_scalar<<<1, 1, 0, stream>>>(Plog[pc], 0.0f);

  for (int c = 0; c < NCHUNK; ++c) {
    // level 1: gather Kraus ops by symbol, Q_b = K[s_{2b+1}] * K[s_{2b}]
    k_cgemm<<<dim3(8, PAIRS_PER_CHUNK), 256, 0, stream>>>(KC, KC, LVA,
        selA + c * PAIRS_PER_CHUNK, 0, 0,
        selB + c * PAIRS_PER_CHUNK, 0, 0, 1, 0, 1, 0);
    k_rescale<<<PAIRS_PER_CHUNK, 256, 0, stream>>>(LVA,
        nullptr, 0, 0, nullptr, 0, 0, logsA);

    float* srcM = LVA; float* dstM = LVB;
    float* srcL = logsA; float* dstL = logsB;
    for (int cnt = PAIRS_PER_CHUNK / 2; cnt >= 1; cnt >>= 1) {
      // dst_b = src_{2b+1} * src_{2b}
      k_cgemm<<<dim3(8, cnt), 256, 0, stream>>>(srcM, srcM, dstM,
          nullptr, 2, 1, nullptr, 2, 0, 1, 0, 1, 0);
      k_rescale<<<cnt, 256, 0, stream>>>(dstM,
          srcL, 2, 0, srcL, 2, 1, dstL);
      float* tmp_m = srcM; srcM = dstM; dstM = tmp_m;
      float* tmp_l = srcL; srcL = dstL; dstL = tmp_l;
    }
    // chunk product in srcM[0], its log scale in srcL[0]
    // combine: P_new = chunkProduct * P_old
    k_cgemm<<<dim3(8, 1), 256, 0, stream>>>(srcM, Pm[pc], Pm[1 - pc],
        nullptr, 0, 0, nullptr, 0, 0, 0, 0, 1, 0);
    k_rescale<<<1, 256, 0, stream>>>(Pm[1 - pc],
        Plog[pc], 0, 0, srcL, 0, 0, Plog[1 - pc]);
    pc ^= 1;
  }

  // -------- final: v = Re tr(P rho0 P^H rhol) --------
  k_conjT<<<dim3(PL / 256, 1), 256, 0, stream>>>(Pm[pc], PH);
  k_cgemm<<<dim3(8, 1), 256, 0, stream>>>(Pm[pc], RHOR, W1,
      nullptr, 0, 0, nullptr, 0, 0, 0, 0, 1, 0);
  k_cgemm<<<dim3(8, 1), 256, 0, stream>>>(W1, PH, W2,
      nullptr, 0, 0, nullptr, 0, 0, 0, 0, 1, 0);
  k_dot_trace<<<1, 256, 0, stream>>>(W2, RHOL, vval);
  k_finalize<<<1, 1, 0, stream>>>(vval, Plog[pc], lam, (float*)d_out, (float)SEQL);
}